// mlp_16621523435686
// MI455X (gfx1250) — compile-verified
//
#include <hip/hip_runtime.h>

// Fused spiking-MLP block for MI455X (gfx1250, wave32):
//   h1 = LN(x @ W1^T + b1); s1 = LIF(h1); h2 = LN(s1 @ W2^T + b2); out = LIF(h2)
// One workgroup = 4 tokens x 4 timesteps = 16-row M-tile, fp32 WMMA 16x16x4.

typedef float v2f __attribute__((ext_vector_type(2)));
typedef float v8f __attribute__((ext_vector_type(8)));

namespace {
constexpr int Bb = 32;    // batch
constexpr int Tt = 4;     // timesteps
constexpr int Ll = 196;   // sequence length
constexpr int Dd = 384;   // model dim
constexpr int Hh = 1536;  // hidden dim
constexpr int XP = Dd + 4;   // LDS pitch (floats), 388 % 64 == 4 -> conflict-free
constexpr int HP = Hh + 4;   // 1540 % 64 == 4
constexpr int YP = Dd + 4;
constexpr float kThr = 1.0f;
constexpr float kEps = 1e-5f;
}

__global__ __launch_bounds__(256)
void snn_mlp_fused(const float* __restrict__ x,
                   const float* __restrict__ W1, const float* __restrict__ b1,
                   const float* __restrict__ g1, const float* __restrict__ be1,
                   const float* __restrict__ W2, const float* __restrict__ b2,
                   const float* __restrict__ g2, const float* __restrict__ be2,
                   float* __restrict__ out)
{
  __shared__ float Xs[16][XP];    // 24.25 KB  : input rows (row r = tok*4 + t)
  __shared__ float H1s[16][HP];   // 96.25 KB  : h1 -> normalized -> spikes (in place)
  __shared__ float Y2s[16][YP];   // 24.25 KB  : h2 -> normalized -> spikes

  const int tid  = threadIdx.x;
  const int lane = tid & 31;           // wave32
  const int wid  = tid >> 5;           // 8 waves
  const int tok0 = blockIdx.x * 4;

  // ---------------- stage X tile: 16 rows x 384, coalesced ----------------
  for (int i = tid; i < 16 * Dd; i += 256) {
    const int r = i / Dd, d = i - r * Dd;
    const int j = r >> 2, t = r & 3;
    const int tok = tok0 + j;
    const int b = tok / Ll, l = tok - b * Ll;
    Xs[r][d] = x[(((size_t)b * Tt + t) * Ll + l) * Dd + d];
  }
  __syncthreads();

  // WMMA f32 16x16x4 fragment mapping (wave32):
  //  A[m][k]: lane = m + 16*(k>=2), vgpr = k%2   -> row = npos, k-offset = 2*nsub
  //  B[k][n]: lane = n + 16*(k>=2), vgpr = k%2
  //  C/D    : vgpr i -> row i (lanes 0-15) / i+8 (lanes 16-31), col = lane%16
  const int nsub = lane >> 4;
  const int npos = lane & 15;
  const int koff = 2 * nsub;

  // ---------------- GEMM1: [16x384] x [384x1536] (+b1) ----------------
  for (int tile = wid; tile < Hh / 16; tile += 8) {
    const int h0 = tile * 16;
    const float* w1p = W1 + (size_t)(h0 + npos) * Dd + koff;  // row h0+npos of W1[H,D]
    __builtin_prefetch(w1p, 0, 1);
    v8f c = {};
    #pragma unroll 4
    for (int k = 0; k < Dd / 4; ++k) {
      v2f a  = *(const v2f*)&Xs[npos][4 * k + koff];          // ds_load_b64
      v2f bm = *(const v2f*)(w1p + 4 * k);                    // global_load_b64 (L2-hot)
      c = __builtin_amdgcn_wmma_f32_16x16x4_f32(false, a, false, bm,
                                                (short)0, c, false, false);
    }
    const float bias = b1[h0 + npos];
    #pragma unroll
    for (int i2 = 0; i2 < 8; ++i2)
      H1s[i2 + 8 * nsub][h0 + npos] = c[i2] + bias;
  }
  __syncthreads();

  // ---------------- LayerNorm over H per row (16 groups x 16 lanes) --------
  {
    const int g = tid >> 4, p = tid & 15;
    float s = 0.f, q = 0.f;
    for (int h = p; h < Hh; h += 16) { float v = H1s[g][h]; s += v; q += v * v; }
    #pragma unroll
    for (int m = 8; m >= 1; m >>= 1) { s += __shfl_xor(s, m, 32); q += __shfl_xor(q, m, 32); }
    const float mu  = s * (1.0f / Hh);
    const float var = q * (1.0f / Hh) - mu * mu;
    const float inv = rsqrtf(var + kEps);
    for (int h = p; h < Hh; h += 16)
      H1s[g][h] = (H1s[g][h] - mu) * inv * g1[h] + be1[h];
  }
  __syncthreads();

  // ---------------- LIF 1: sequential over T per (token, h) chain ----------
  for (int i = tid; i < 4 * Hh; i += 256) {
    const int j = i / Hh, h = i - j * Hh;
    float v = 0.f;
    #pragma unroll
    for (int t = 0; t < 4; ++t) {
      const float xx = H1s[j * 4 + t][h];
      v = 0.5f * (v + xx);                      // v += (x - v)/TAU, TAU=2
      const float sp = (v >= kThr) ? 1.0f : 0.0f;
      H1s[j * 4 + t][h] = sp;                   // spikes replace h1 in place
      v *= (1.0f - sp);                         // hard reset
    }
  }
  __syncthreads();

  // ---------------- GEMM2: [16x1536] x [1536x384] (+b2) ----------------
  for (int tile = wid; tile < Dd / 16; tile += 8) {
    const int d0 = tile * 16;
    const float* w2p = W2 + (size_t)(d0 + npos) * Hh + koff;  // row d0+npos of W2[D,H]
    __builtin_prefetch(w2p, 0, 1);
    v8f c = {};
    #pragma unroll 4
    for (int k = 0; k < Hh / 4; ++k) {
      v2f a  = *(const v2f*)&H1s[npos][4 * k + koff];
      v2f bm = *(const v2f*)(w2p + 4 * k);
      c = __builtin_amdgcn_wmma_f32_16x16x4_f32(false, a, false, bm,
                                                (short)0, c, false, false);
    }
    const float bias = b2[d0 + npos];
    #pragma unroll
    for (int i2 = 0; i2 < 8; ++i2)
      Y2s[i2 + 8 * nsub][d0 + npos] = c[i2] + bias;
  }
  __syncthreads();

  // ---------------- LayerNorm over D per row ----------------
  {
    const int g = tid >> 4, p = tid & 15;
    float s = 0.f, q = 0.f;
    for (int d = p; d < Dd; d += 16) { float v = Y2s[g][d]; s += v; q += v * v; }
    #pragma unroll
    for (int m = 8; m >= 1; m >>= 1) { s += __shfl_xor(s, m, 32); q += __shfl_xor(q, m, 32); }
    const float mu  = s * (1.0f / Dd);
    const float var = q * (1.0f / Dd) - mu * mu;
    const float inv = rsqrtf(var + kEps);
    for (int d = p; d < Dd; d += 16)
      Y2s[g][d] = (Y2s[g][d] - mu) * inv * g2[d] + be2[d];
  }
  __syncthreads();

  // ---------------- LIF 2 ----------------
  for (int i = tid; i < 4 * Dd; i += 256) {
    const int j = i / Dd, d = i - j * Dd;
    float v = 0.f;
    #pragma unroll
    for (int t = 0; t < 4; ++t) {
      const float xx = Y2s[j * 4 + t][d];
      v = 0.5f * (v + xx);
      const float sp = (v >= kThr) ? 1.0f : 0.0f;
      Y2s[j * 4 + t][d] = sp;
      v *= (1.0f - sp);
    }
  }
  __syncthreads();

  // ---------------- store, folding [T,B,L,D] -> [B,T,L,D] transpose --------
  for (int i = tid; i < 16 * Dd; i += 256) {
    const int r = i / Dd, d = i - r * Dd;
    const int j = r >> 2, t = r & 3;
    const int tok = tok0 + j;
    const int b = tok / Ll, l = tok - b * Ll;
    out[(((size_t)b * Tt + t) * Ll + l) * Dd + d] = Y2s[r][d];
  }
}

extern "C" void kernel_launch(void* const* d_in, const int* in_sizes, int n_in,
                              void* d_out, int out_size, void* d_ws, size_t ws_size,
                              hipStream_t stream) {
  (void)in_sizes; (void)n_in; (void)out_size; (void)d_ws; (void)ws_size;
  const float* x   = (const float*)d_in[0];
  const float* W1  = (const float*)d_in[1];
  const float* b1  = (const float*)d_in[2];
  const float* g1  = (const float*)d_in[3];
  const float* be1 = (const float*)d_in[4];
  const float* W2  = (const float*)d_in[5];
  const float* b2  = (const float*)d_in[6];
  const float* g2  = (const float*)d_in[7];
  const float* be2 = (const float*)d_in[8];
  float* out = (float*)d_out;

  const int n_wg = (Bb * Ll) / 4;  // 1568 workgroups, 4 tokens each
  snn_mlp_fused<<<dim3(n_wg), dim3(256), 0, stream>>>(
      x, W1, b1, g1, be1, W2, b2, g2, be2, out);
}